// SAGE_pyg_17119739641947
// MI455X (gfx1250) — compile-verified
//
#include <hip/hip_runtime.h>
#include <hip/hip_bf16.h>

// ---------------------------------------------------------------------------
// GraphSAGE 2-layer + link prediction for MI455X (gfx1250, wave32).
// GEMMs go through V_WMMA_F32_16X16X4_F32 (fp32 in/out — precision-exact vs
// the fp32 reference; GEMM FLOPs (~13 GF) are negligible next to the ~1 GB of
// edge scatter/gather traffic, so fp32 WMMA is the right precision choice).
// ---------------------------------------------------------------------------

typedef __attribute__((ext_vector_type(2))) float v2f;
typedef __attribute__((ext_vector_type(8))) float v8f;

#define D_IN  128
#define D_HID 256

// ---------------------------------------------------------------------------
// utility: zero a float buffer (grid-stride)
// ---------------------------------------------------------------------------
__global__ void zero_f32(float* __restrict__ p, long n) {
    long i = (long)blockIdx.x * blockDim.x + threadIdx.x;
    long stride = (long)gridDim.x * blockDim.x;
    for (; i < n; i += stride) p[i] = 0.0f;
}

// ---------------------------------------------------------------------------
// degree: deg[dst[e]] += 1
// ---------------------------------------------------------------------------
__global__ void degree_kernel(const int* __restrict__ dst, float* __restrict__ deg,
                              int n_edges) {
    int e = blockIdx.x * blockDim.x + threadIdx.x;
    if (e < n_edges) atomicAdd(&deg[dst[e]], 1.0f);
}

// ---------------------------------------------------------------------------
// scatter-add messages: agg[dst[e], :] += x[src[e], :]   (one block per edge)
// ---------------------------------------------------------------------------
__global__ void scatter_add_kernel(const int* __restrict__ src, const int* __restrict__ dst,
                                   const float* __restrict__ x, float* __restrict__ agg,
                                   int n_edges, int d) {
    int e = blockIdx.x;
    if (e >= n_edges) return;
    long s = (long)src[e] * d;
    long t = (long)dst[e] * d;
    for (int f = threadIdx.x; f < d; f += blockDim.x)
        atomicAdd(&agg[t + f], x[s + f]);
}

// ---------------------------------------------------------------------------
// mean: agg[i,f] /= max(deg[i], 1)
// ---------------------------------------------------------------------------
__global__ void mean_div_kernel(float* __restrict__ agg, const float* __restrict__ deg,
                                long total, int d) {
    long i = (long)blockIdx.x * blockDim.x + threadIdx.x;
    if (i < total) agg[i] *= __frcp_rn(fmaxf(deg[i / d], 1.0f));
}

// ---------------------------------------------------------------------------
// Fused dual-GEMM via V_WMMA_F32_16X16X4_F32:
//   out = act(A1 @ W1 + A2 @ W2 + bias)
// A1,A2: [nrows x K] row-major;  W1,W2: [K x ncols] row-major;  out: [nrows x ncols]
// One wave computes one 16x16 tile.  blockDim.x = 256 (8 waves -> 8 col tiles).
// nrows % 16 == 0 (50000 = 16*3125), K % 4 == 0, ncols % 16 == 0: no tails,
// EXEC stays all-ones as required by WMMA.
//
// Fragment layouts per CDNA5 ISA 7.12.2 (fp32, wave32):
//  A 16x4:  lanes 0-15 -> M=lane, {v0,v1}={K0,K1}; lanes 16-31 -> {K2,K3}
//  B 4x16:  v0: lanes0-15 K=0 / lanes16-31 K=2;  v1: K=1 / K=3; lane&15 = N
//  C 16x16: v in 0..7: lanes0-15 -> M=v, lanes16-31 -> M=v+8; lane&15 = N
// ---------------------------------------------------------------------------
__global__ __launch_bounds__(256)
void wmma_dual_gemm_kernel(const float* __restrict__ A1, const float* __restrict__ W1,
                           const float* __restrict__ A2, const float* __restrict__ W2,
                           const float* __restrict__ bias, float* __restrict__ out,
                           int K, int ncols, int do_relu) {
    const int lane = threadIdx.x & 31;
    const int wave = threadIdx.x >> 5;
    const int half = lane >> 4;        // 0: lanes 0-15, 1: lanes 16-31
    const int l16  = lane & 15;

    const int row0 = blockIdx.x * 16;
    const int col0 = (blockIdx.y * (blockDim.x >> 5) + wave) * 16;
    if (col0 >= ncols) return;

    const long arow = (long)(row0 + l16) * K;   // A row for this lane
    v8f c = {};

    for (int k0 = 0; k0 < K; k0 += 4) {
        const int ka = k0 + half * 2;           // this lane-half's K base

        v2f a1, b1, a2, b2;
        a1.x = A1[arow + ka];
        a1.y = A1[arow + ka + 1];
        b1.x = W1[(long)ka * ncols + col0 + l16];
        b1.y = W1[(long)(ka + 1) * ncols + col0 + l16];

        a2.x = A2[arow + ka];
        a2.y = A2[arow + ka + 1];
        b2.x = W2[(long)ka * ncols + col0 + l16];
        b2.y = W2[(long)(ka + 1) * ncols + col0 + l16];

        // v_wmma_f32_16x16x4_f32 : D = A*B + C
        c = __builtin_amdgcn_wmma_f32_16x16x4_f32(false, a1, false, b1,
                                                  (short)0, c, false, false);
        c = __builtin_amdgcn_wmma_f32_16x16x4_f32(false, a2, false, b2,
                                                  (short)0, c, false, false);
    }

    const float bb = bias[col0 + l16];
#pragma unroll
    for (int v = 0; v < 8; ++v) {
        const int m = v + half * 8;
        float val = c[v] + bb;
        if (do_relu) val = fmaxf(val, 0.0f);
        out[(long)(row0 + m) * ncols + col0 + l16] = val;
    }
}

// ---------------------------------------------------------------------------
// pair scoring: out[p] = sigmoid( h2[m0]·Wlin[0:128] + h2[m1]·Wlin[128:256] + blin )
// one wave per pair, shuffle reduction (wave32).
// ---------------------------------------------------------------------------
__global__ void pair_score_kernel(const float* __restrict__ h2, const int* __restrict__ mask,
                                  const float* __restrict__ Wlin, const float* __restrict__ blin,
                                  float* __restrict__ out, int n_pairs) {
    const int lane = threadIdx.x & 31;
    const int pair = blockIdx.x * (blockDim.x >> 5) + (threadIdx.x >> 5);
    if (pair >= n_pairs) return;

    const long m0 = (long)mask[2 * pair]     * D_IN;
    const long m1 = (long)mask[2 * pair + 1] * D_IN;

    float acc = 0.0f;
#pragma unroll
    for (int f = lane; f < D_IN; f += 32) {
        acc += h2[m0 + f] * Wlin[f];
        acc += h2[m1 + f] * Wlin[D_IN + f];
    }
#pragma unroll
    for (int off = 16; off > 0; off >>= 1)
        acc += __shfl_down(acc, off, 32);

    if (lane == 0) {
        const float z = acc + blin[0];
        out[pair] = 1.0f / (1.0f + __expf(-z));
    }
}

// ---------------------------------------------------------------------------
// launch
// ---------------------------------------------------------------------------
extern "C" void kernel_launch(void* const* d_in, const int* in_sizes, int n_in,
                              void* d_out, int out_size, void* d_ws, size_t ws_size,
                              hipStream_t stream) {
    const float* features = (const float*)d_in[0];   // [N, 128]
    const int*   edge     = (const int*)  d_in[1];   // [2, E]
    const int*   mask     = (const int*)  d_in[2];   // [P, 2]
    const float* W1l      = (const float*)d_in[3];   // [128, 256]
    const float* b1l      = (const float*)d_in[4];   // [256]
    const float* W1r      = (const float*)d_in[5];   // [128, 256]
    const float* W2l      = (const float*)d_in[6];   // [256, 128]
    const float* b2l      = (const float*)d_in[7];   // [128]
    const float* W2r      = (const float*)d_in[8];   // [256, 128]
    const float* Wlin     = (const float*)d_in[9];   // [256, 1]
    const float* blin     = (const float*)d_in[10];  // [1]
    float*       outp     = (float*)d_out;           // [P, 1]

    const int N = in_sizes[0] / D_IN;
    const int E = in_sizes[1] / 2;
    const int P = in_sizes[2] / 2;

    const int* e_src = edge;
    const int* e_dst = edge + E;

    // workspace partition
    float* ws   = (float*)d_ws;
    float* deg  = ws;                 // N
    float* agg1 = deg  + N;           // N*128
    float* h    = agg1 + (long)N * D_IN;   // N*256
    float* agg2 = h    + (long)N * D_HID;  // N*256
    float* h2   = agg2 + (long)N * D_HID;  // N*128
    (void)ws_size; (void)n_in; (void)out_size;

    const int ZB = 256;
    const int row_tiles = N / 16;     // 50000/16 = 3125 exact

    // ---- layer 1: mean aggregation ----
    zero_f32<<<2048, ZB, 0, stream>>>(deg, (long)N);
    zero_f32<<<4096, ZB, 0, stream>>>(agg1, (long)N * D_IN);
    degree_kernel<<<(E + ZB - 1) / ZB, ZB, 0, stream>>>(e_dst, deg, E);
    scatter_add_kernel<<<E, 128, 0, stream>>>(e_src, e_dst, features, agg1, E, D_IN);
    {
        long tot = (long)N * D_IN;
        mean_div_kernel<<<(int)((tot + ZB - 1) / ZB), ZB, 0, stream>>>(agg1, deg, tot, D_IN);
    }
    // h = relu(agg1 @ W1l + b1l + features @ W1r)   [N, 256]
    {
        dim3 grid(row_tiles, D_HID / 16 / 8);   // 8 waves/block, each 1 col tile
        wmma_dual_gemm_kernel<<<grid, 256, 0, stream>>>(agg1, W1l, features, W1r,
                                                        b1l, h, D_IN, D_HID, 1);
    }

    // ---- layer 2: mean aggregation on h ----
    zero_f32<<<8192, ZB, 0, stream>>>(agg2, (long)N * D_HID);
    scatter_add_kernel<<<E, 256, 0, stream>>>(e_src, e_dst, h, agg2, E, D_HID);
    {
        long tot = (long)N * D_HID;
        mean_div_kernel<<<(int)((tot + ZB - 1) / ZB), ZB, 0, stream>>>(agg2, deg, tot, D_HID);
    }
    // h2 = agg2 @ W2l + b2l + h @ W2r   [N, 128]
    {
        dim3 grid(row_tiles, D_IN / 16 / 8);    // 8 col tiles -> grid.y = 1
        wmma_dual_gemm_kernel<<<grid, 256, 0, stream>>>(agg2, W2l, h, W2r,
                                                        b2l, h2, D_HID, D_IN, 0);
    }

    // ---- pair scoring ----
    {
        const int waves_per_block = 256 / 32;   // 8 pairs per block
        const int blocks = (P + waves_per_block - 1) / waves_per_block;
        pair_score_kernel<<<blocks, 256, 0, stream>>>(h2, mask, Wlin, blin, outp, P);
    }
}